// BlockDiagonalDense_13958643712272
// MI455X (gfx1250) — compile-verified
//
#include <hip/hip_runtime.h>

// BlockDiagonalDense for MI455X (gfx1250, wave32).
// y[..,h,b,t] = sum_s x[..,h,b,s] * K[h,b,s,t] + bias[h,b,t]
// Memory-bound (2 FLOP/byte): stream x/y once with nontemporal b128 loads /
// b32 stores; math via chained V_WMMA_F32_16X16X4_F32 on a block-diagonal
// 16x16 B built from two adjacent 8x8 kernel blocks.

typedef __attribute__((ext_vector_type(2))) float v2f;
typedef __attribute__((ext_vector_type(4))) float v4f;
typedef __attribute__((ext_vector_type(8))) float v8f;

#define FEATURES        4096
#define TOK_TILE        16          // tokens per WMMA tile (M)
#define WAVES_PER_BLOCK 8
#define TILES_PER_WAVE  4

// lane L <-> lane L^16 across the wave32 (ds_swizzle group-of-32, xor=0x10)
__device__ __forceinline__ float swap16(float v) {
    return __int_as_float(__builtin_amdgcn_ds_swizzle(__float_as_int(v), 0x401f));
}

__global__ __launch_bounds__(256)
void blockdiag_wmma_kernel(const float* __restrict__ x,
                           const float* __restrict__ w,     // [16][32][8][8]
                           const float* __restrict__ bias,  // [16][32][8]
                           float* __restrict__ y,
                           int num_token_tiles)
{
    const int lane = threadIdx.x & 31;
    const int wave = threadIdx.x >> 5;
    const int p    = blockIdx.x;        // feature-pair tile: features [16p, 16p+16)
    const int f0   = p << 4;
    const int n    = lane & 15;         // column within the 16-wide tile
    const int hi   = lane >> 4;         // 0: lanes 0-15, 1: lanes 16-31

    // ---- Build B: block-diagonal 16x16 from kernels (h, 2b') and (h, 2b'+1).
    // Chunk j covers K rows 4j..4j+3. Register r of chunk j holds:
    //   lanes 0-15: B[4j+r][n]   lanes 16-31: B[4j+r+2][n]
    v2f B[4];
    #pragma unroll
    for (int j = 0; j < 4; ++j) {
        #pragma unroll
        for (int r = 0; r < 2; ++r) {
            const int k   = 4 * j + r + 2 * hi;          // 0..15
            const int blk = 2 * p + (k >> 3);            // flat (h*32+b) index
            const float v = w[blk * 64 + (k & 7) * 8 + (n & 7)];
            B[j][r] = ((k >> 3) == (n >> 3)) ? v : 0.0f; // zero off-diagonal
        }
    }
    const float bias_v = bias[(p << 4) + n];             // depends only on column

    const int wave_glb = blockIdx.y * WAVES_PER_BLOCK + wave;

    #pragma unroll
    for (int i = 0; i < TILES_PER_WAVE; ++i) {
        const int tile = wave_glb * TILES_PER_WAVE + i;  // wave-uniform
        if (tile >= num_token_tiles) return;

        // Each lane loads 8 consecutive features (two b128) of one token row:
        // lanes 0-15: feats f0+0..7 of row n; lanes 16-31: feats f0+8..15 of row n.
        const size_t row = (size_t)tile * TOK_TILE + n;
        const float* src = x + row * FEATURES + f0 + hi * 8;
        const v4f xa = __builtin_nontemporal_load((const v4f*)src);
        const v4f xb = __builtin_nontemporal_load((const v4f*)(src + 4));

        // Half-wave exchange to assemble the four 16x4 A fragments.
        const float sax = swap16(xa.x), say = swap16(xa.y);
        const float saz = swap16(xa.z), saw = swap16(xa.w);
        const float sbx = swap16(xb.x), sby = swap16(xb.y);
        const float sbz = swap16(xb.z), sbw = swap16(xb.w);

        const v2f A0 = hi ? (v2f){saz, saw} : (v2f){xa.x, xa.y}; // K 0..3
        const v2f A1 = hi ? (v2f){sbz, sbw} : (v2f){xb.x, xb.y}; // K 4..7
        const v2f A2 = hi ? (v2f){xa.z, xa.w} : (v2f){sax, say}; // K 8..11
        const v2f A3 = hi ? (v2f){xb.z, xb.w} : (v2f){sbx, sby}; // K 12..15

        v8f acc;
        #pragma unroll
        for (int r = 0; r < 8; ++r) acc[r] = bias_v;

        acc = __builtin_amdgcn_wmma_f32_16x16x4_f32(false, A0, false, B[0],
                                                    (short)0, acc, false, false);
        acc = __builtin_amdgcn_wmma_f32_16x16x4_f32(false, A1, false, B[1],
                                                    (short)0, acc, false, false);
        acc = __builtin_amdgcn_wmma_f32_16x16x4_f32(false, A2, false, B[2],
                                                    (short)0, acc, false, false);
        acc = __builtin_amdgcn_wmma_f32_16x16x4_f32(false, A3, false, B[3],
                                                    (short)0, acc, false, false);

        // D layout: VGPR r -> row (tile*16 + r + 8*hi), col f0 + n.
        float* dst = y + ((size_t)tile * TOK_TILE + 8 * hi) * FEATURES + f0 + n;
        #pragma unroll
        for (int r = 0; r < 8; ++r)
            __builtin_nontemporal_store(acc[r], dst + (size_t)r * FEATURES);
    }
}

extern "C" void kernel_launch(void* const* d_in, const int* in_sizes, int n_in,
                              void* d_out, int out_size, void* d_ws, size_t ws_size,
                              hipStream_t stream) {
    const float* x    = (const float*)d_in[0];
    const float* w    = (const float*)d_in[1];
    const float* bias = (const float*)d_in[2];
    float* y = (float*)d_out;

    const long tokens       = (long)in_sizes[0] / FEATURES;   // 16384
    const int  token_tiles  = (int)(tokens / TOK_TILE);       // 1024
    const int  pair_tiles   = FEATURES / 16;                  // 256

    int gy = token_tiles / (WAVES_PER_BLOCK * TILES_PER_WAVE);
    if (gy < 1) gy = 1;

    dim3 grid(pair_tiles, gy);
    blockdiag_wmma_kernel<<<grid, 256, 0, stream>>>(x, w, bias, y, token_tiles);
}